// ARX_58935541235970
// MI455X (gfx1250) — compile-verified
//
#include <hip/hip_runtime.h>

typedef __attribute__((ext_vector_type(2))) float v2f;
typedef __attribute__((ext_vector_type(4))) float v4f;
typedef __attribute__((ext_vector_type(8))) float v8f;

#define AR        8
#define SEQ       4096
#define BATCH     8192
#define OUTW      (AR + SEQ)      /* 4104 */
#define BLK_STEPS 16
#define NBLK      (SEQ / BLK_STEPS) /* 256 */

// Build P (16x8, row-major) in workspace: pred_{t} = sum_j P[t][j] * s_j,
// where s_j = carry[:, j] (s_0 oldest). Derived by iterating the companion
// recurrence on coefficient vectors. One thread; cost is negligible.
__global__ void arx_build_P(const float* __restrict__ W, float* __restrict__ P) {
    if (threadIdx.x != 0 || blockIdx.x != 0) return;
    float w[AR];
#pragma unroll
    for (int i = 0; i < AR; ++i) w[i] = W[i];

    float win[AR][AR]; // win[i][j] = coeff of s_j in window element x_{t+i}
#pragma unroll
    for (int i = 0; i < AR; ++i)
#pragma unroll
        for (int j = 0; j < AR; ++j) win[i][j] = (i == j) ? 1.0f : 0.0f;

    for (int t = 0; t < BLK_STEPS; ++t) {
        float c[AR];
#pragma unroll
        for (int j = 0; j < AR; ++j) {
            float acc = 0.0f;
#pragma unroll
            for (int i = 0; i < AR; ++i) acc += w[i] * win[i][j];
            c[j] = acc;
            P[t * AR + j] = acc;
        }
        // shift window: x_{t+1+i} = x_{t+i+1}, new last element = pred_t
#pragma unroll
        for (int i = 0; i < AR - 1; ++i)
#pragma unroll
            for (int j = 0; j < AR; ++j) win[i][j] = win[i + 1][j];
#pragma unroll
        for (int j = 0; j < AR; ++j) win[AR - 1][j] = c[j];
    }
}

// One wave per 16-row batch tile. A = P (loop-invariant), B = state (8x16),
// D = 16 preds x 16 batch cols. Next state = last 8 preds = upper lane halves
// of D VGPRs 0..7 -> ds_swizzle SWAPX16 + cndmask into next B operands.
__global__ __launch_bounds__(32) void arx_wmma_kernel(const float* __restrict__ y,
                                                      const float* __restrict__ P,
                                                      float* __restrict__ out) {
    const int lane = threadIdx.x & 31;
    const int tile = blockIdx.x;           // 512 tiles of 16 batch rows
    const int base = tile * 16;
    const int col  = lane & 15;            // batch col within tile / P row (M)
    const bool hi  = lane >= 16;
    const int kb   = hi ? 2 : 0;           // K pair held by this lane half

    // A operands: a0 covers K=0..3 of P, a1 covers K=4..7.
    // VGPR0 = {K=kb}, VGPR1 = {K=kb+1} per 16x4 f32 A layout.
    v2f a0 = *(const v2f*)(P + col * AR + kb);
    v2f a1 = *(const v2f*)(P + col * AR + 4 + kb);

    // Initial B from y: B is 8x16 (state-index K x batch col N).
    const float* yrow = y + (size_t)(base + col) * AR;
    v2f b0 = *(const v2f*)(yrow + kb);      // {s_kb, s_kb+1}
    v2f b1 = *(const v2f*)(yrow + 4 + kb);  // {s_kb+4, s_kb+5}

    // out[:, 0:8] = y (lanes 0-15 write cols 0..3, lanes 16-31 cols 4..7)
    float* orow = out + (size_t)(base + col) * OUTW;
    {
        v4f yc = *(const v4f*)(yrow + (hi ? 4 : 0));
        *(v4f*)(orow + (hi ? 4 : 0)) = yc;
    }

#pragma unroll 4
    for (int blk = 0; blk < NBLK; ++blk) {
        v8f acc = {0.f, 0.f, 0.f, 0.f, 0.f, 0.f, 0.f, 0.f};
        acc = __builtin_amdgcn_wmma_f32_16x16x4_f32(false, a0, false, b0,
                                                    (short)0, acc, false, false);
        acc = __builtin_amdgcn_wmma_f32_16x16x4_f32(false, a1, false, b1,
                                                    (short)0, acc, false, false);

        // Per-lane 8 contiguous preds: lanes<16 hold steps t0..t0+7 of col,
        // lanes>=16 hold steps t0+8..t0+15 of col. Streaming (NT) stores.
        float* dst = orow + AR + blk * BLK_STEPS + (hi ? 8 : 0);
        v4f lo = { acc[0], acc[1], acc[2], acc[3] };
        v4f hi4 = { acc[4], acc[5], acc[6], acc[7] };
        __builtin_nontemporal_store(lo,  (v4f*)dst);
        __builtin_nontemporal_store(hi4, (v4f*)(dst + 4));

        // New state s'_k = pred_{t0+8+k} = D VGPR k, upper lane half.
        // lanes<16 fetch lane+16 (SWAPX16); lanes>=16 take VGPR k+2 in place.
        float sw0 = __int_as_float(__builtin_amdgcn_ds_swizzle(__float_as_int(acc[0]), 0x401F));
        float sw1 = __int_as_float(__builtin_amdgcn_ds_swizzle(__float_as_int(acc[1]), 0x401F));
        float sw4 = __int_as_float(__builtin_amdgcn_ds_swizzle(__float_as_int(acc[4]), 0x401F));
        float sw5 = __int_as_float(__builtin_amdgcn_ds_swizzle(__float_as_int(acc[5]), 0x401F));
        b0.x = hi ? acc[2] : sw0;   // K=0 / K=2
        b0.y = hi ? acc[3] : sw1;   // K=1 / K=3
        b1.x = hi ? acc[6] : sw4;   // K=4 / K=6
        b1.y = hi ? acc[7] : sw5;   // K=5 / K=7
    }
}

extern "C" void kernel_launch(void* const* d_in, const int* in_sizes, int n_in,
                              void* d_out, int out_size, void* d_ws, size_t ws_size,
                              hipStream_t stream) {
    const float* y = (const float*)d_in[0];   // (8192, 8)
    // d_in[1] = u (8192, 4096): unused by the reference recurrence
    const float* W = (const float*)d_in[2];   // (1, 9); only first 8 used
    float* out = (float*)d_out;               // (8192, 4104)
    float* P = (float*)d_ws;                  // 16x8 floats = 512 bytes

    arx_build_P<<<1, 32, 0, stream>>>(W, P);
    arx_wmma_kernel<<<BATCH / 16, 32, 0, stream>>>(y, P, out);
}